// SpeDCT_29850022707953
// MI455X (gfx1250) — compile-verified
//
#include <hip/hip_runtime.h>
#include <math.h>

// Problem constants (from reference setup_inputs)
#define B_  16
#define C_  28
#define H_  256
#define W_  256
#define HW_ (H_ * W_)

#define WAVES_PER_BLOCK 4
#define PIX_PER_WAVE    16
#define TILES_PER_WAVE  4     // each wave handles 4 x 16 = 64 pixels; block covers W=256

typedef __attribute__((ext_vector_type(2))) float v2f;
typedef __attribute__((ext_vector_type(8))) float v8f;

// Table layout in d_ws: v2f tab[2 /*mat*/][2 /*mtile*/][7 /*j*/][32 /*lane*/]
#define TAB_ENTRIES (2 * 2 * 7 * 32)   // 896 float2 = 7168 bytes

// V_WMMA_F32_16X16X4_F32: D(16x16 f32, 8 VGPR) = A(16x4 f32, 2 VGPR) x B(4x16 f32, 2 VGPR) + C
static __device__ __forceinline__ v8f wmma_f32(v2f a, v2f b, v8f c) {
    return __builtin_amdgcn_wmma_f32_16x16x4_f32(
        /*neg_a=*/false, a, /*neg_b=*/false, b,
        /*c_mod=*/(short)0, c, /*reuse_a=*/false, /*reuse_b=*/false);
}

// ---------------------------------------------------------------------------
// Init kernel: build DCT / inverse-DCT A-operand tables in per-lane layout.
//   A layout (f32 16x4): component v, lane l: M = mt*16 + (l&15), K = 4j + v + 2*(l>>4)
//   mat 0 (GEMM1, D):    D[k=M][n=K]    = 2*cos(pi*(2K+1)*M / (2C)), 0 for M>=C
//   mat 1 (GEMM2, Dinv): Dinv[n=M][k=K] = cos(pi*(2M+1)*K / (2C)) * ((K==0)?0.5:1)/C
// ---------------------------------------------------------------------------
__global__ void init_dct_tables(float2* __restrict__ tab) {
    int idx = blockIdx.x * blockDim.x + threadIdx.x;
    if (idx >= TAB_ENTRIES) return;
    int lane = idx & 31;
    int j    = (idx >> 5) % 7;
    int mt   = ((idx >> 5) / 7) & 1;
    int mat  = idx / (32 * 7 * 2);

    int lm = lane & 15, lh = lane >> 4;
    int M = mt * 16 + lm;
    const float PI = 3.14159265358979323846f;
    float2 r;
    float* rv = &r.x;
    for (int v = 0; v < 2; ++v) {
        int K = 4 * j + v + 2 * lh;          // 0..27
        float val;
        if (mat == 0) {
            float ang = PI * (2.0f * (float)K + 1.0f) * (float)M / (2.0f * (float)C_);
            val = (M < C_) ? 2.0f * cosf(ang) : 0.0f;   // zero pad rows k >= 28
        } else {
            float ang = PI * (2.0f * (float)M + 1.0f) * (float)K / (2.0f * (float)C_);
            val = cosf(ang) * (((K == 0) ? 0.5f : 1.0f) * (1.0f / (float)C_));
        }
        rv[v] = val;
    }
    tab[idx] = r;
}

// ---------------------------------------------------------------------------
// Main kernel: one block per (b, h); 4 waves x 4 tiles cover all 256 pixels.
// ---------------------------------------------------------------------------
__global__ __launch_bounds__(WAVES_PER_BLOCK * 32)
void spedct_kernel(const float* __restrict__ x,
                   const float* __restrict__ wgt,
                   float* __restrict__ out,
                   const v2f* __restrict__ tab)
{
    const int lane = threadIdx.x & 31;
    const int wave = threadIdx.x >> 5;
    const int lm   = lane & 15;
    const int lh   = lane >> 4;

    const int b = blockIdx.x / H_;
    const int h = blockIdx.x % H_;

    // ---- Load A operands (coalesced b64, L2-resident 7KB table) ----------
    v2f A1[2][7], A2[2][7];
#pragma unroll
    for (int mt = 0; mt < 2; ++mt) {
#pragma unroll
        for (int j = 0; j < 7; ++j) {
            A1[mt][j] = tab[((0 * 2 + mt) * 7 + j) * 32 + lane];
            A2[mt][j] = tab[((1 * 2 + mt) * 7 + j) * 32 + lane];
        }
    }

    const float* xrow = x   + (size_t)b * C_ * HW_ + (size_t)h * W_;
    float*       orow = out + (size_t)b * C_ * HW_ + (size_t)h * W_;
    const float* wrow = wgt + (size_t)h * W_ * C_;

    __shared__ float lds[WAVES_PER_BLOCK][32 * 16];
    float* Lw = &lds[wave][0];

    for (int t = 0; t < TILES_PER_WAVE; ++t) {
        const int w0 = wave * (TILES_PER_WAVE * PIX_PER_WAVE) + t * PIX_PER_WAVE;

        // ---- GEMM1: Y[k,w] = sum_n D[k,n] * x[b,n,h,w0+w] ----------------
        // B layout (f32 4x16): comp v, lane l: K = 4j + v + 2*lh, N = lm
        const float* xb = xrow + w0;
        v8f acc0 = {};
        v8f acc1 = {};
#pragma unroll
        for (int j = 0; j < 7; ++j) {
            int n0 = 4 * j + 2 * lh;
            v2f Bx;
            Bx[0] = xb[(size_t)(n0 + 0) * HW_ + lm];
            Bx[1] = xb[(size_t)(n0 + 1) * HW_ + lm];
            acc0 = wmma_f32(A1[0][j], Bx, acc0);
            acc1 = wmma_f32(A1[1][j], Bx, acc1);
        }

        // ---- Spectral scale: Y[k,w] *= weight[h, w0+w, k] ----------------
        // C/D layout: VGPR r, lane l: M = r + 8*lh (+16 for acc1), N = lm
        // acc1 rows k >= 28 are exactly zero (A1 rows zeroed), so an
        // index-clamped (in-bounds, finite) weight is equivalent to masking
        // and avoids divergent exec-masked loads.
        const float* wp = wrow + (size_t)(w0 + lm) * C_;
#pragma unroll
        for (int r = 0; r < 8; ++r) {
            int k0 = r + 8 * lh;                 // 0..15, always valid
            acc0[r] *= wp[k0];
            int k1 = k0 + 16;                    // 16..31
            int k1c = (k1 > C_ - 1) ? (C_ - 1) : k1;
            acc1[r] *= wp[k1c];
        }

        // ---- Transpose through per-wave LDS tile (32 rows x 16 cols) -----
#pragma unroll
        for (int r = 0; r < 8; ++r) {
            Lw[(r + 8 * lh) * 16 + lm]      = acc0[r];
            Lw[(r + 8 * lh + 16) * 16 + lm] = acc1[r];
        }
        asm volatile("" ::: "memory");  // keep DS stores before DS loads (DS is in-order per wave)

        // ---- GEMM2: Z[n,w] = sum_k Dinv[n,k] * Y'[k,w] -------------------
        v8f acc2_0 = {};
        v8f acc2_1 = {};
#pragma unroll
        for (int j = 0; j < 7; ++j) {
            int k0 = 4 * j + 2 * lh;
            v2f By;
            By[0] = Lw[(k0 + 0) * 16 + lm];
            By[1] = Lw[(k0 + 1) * 16 + lm];
            acc2_0 = wmma_f32(A2[0][j], By, acc2_0);
            acc2_1 = wmma_f32(A2[1][j], By, acc2_1);
        }
        asm volatile("" ::: "memory");  // keep this tile's DS loads before next tile's DS stores

        // ---- Store out[b, n, h, w0+w] for n < 28 (streaming, NT hint) ----
        float* ob = orow + w0 + lm;
#pragma unroll
        for (int r = 0; r < 8; ++r) {
            int n0 = r + 8 * lh;
            __builtin_nontemporal_store(acc2_0[r], &ob[(size_t)n0 * HW_]);
            int n1 = n0 + 16;
            if (n1 < C_)
                __builtin_nontemporal_store(acc2_1[r], &ob[(size_t)n1 * HW_]);
        }
    }
}

extern "C" void kernel_launch(void* const* d_in, const int* in_sizes, int n_in,
                              void* d_out, int out_size, void* d_ws, size_t ws_size,
                              hipStream_t stream) {
    (void)in_sizes; (void)n_in; (void)out_size; (void)ws_size;
    const float* x   = (const float*)d_in[0];   // [16, 28, 256, 256] f32
    const float* wgt = (const float*)d_in[1];   // [256, 256, 28] f32
    float* out = (float*)d_out;                 // [16, 28, 256, 256] f32

    // 1) Build DCT operand tables in workspace (7168 bytes), every call (deterministic).
    float2* tab = (float2*)d_ws;
    init_dct_tables<<<(TAB_ENTRIES + 127) / 128, 128, 0, stream>>>(tab);

    // 2) Main fused DCT -> scale -> iDCT kernel.
    dim3 grid(B_ * H_);                       // 4096 blocks: one (b, h) row each
    dim3 block(WAVES_PER_BLOCK * 32);         // 128 threads = 4 waves
    spedct_kernel<<<grid, block, 0, stream>>>(x, wgt, out, (const v2f*)tab);
}